// VSMLMV3_2_63471026700804
// MI455X (gfx1250) — compile-verified
//
#include <hip/hip_runtime.h>
#include <hip/hip_bf16.h>
#include <math.h>
#include <stdint.h>

typedef float v2f __attribute__((ext_vector_type(2)));
typedef float v8f __attribute__((ext_vector_type(8)));

#define LSEQ   4096
#define DMODEL 512
#define NHEAD  8

// ----------------------------------------------------------------------
// CDNA5 async global->LDS staging (ASYNCcnt), with manual fallback.
// Probe-confirmed signature (from hipcc diagnostic):
//   __builtin_amdgcn_global_load_async_to_lds_b128(
//       int __vector_size__(16) AS(1)* src, int __vector_size__(16) AS(3)* dst,
//       imm int offset, imm int cpol)
// ----------------------------------------------------------------------
#if __has_builtin(__builtin_amdgcn_global_load_async_to_lds_b128) && \
    __has_builtin(__builtin_amdgcn_s_wait_asynccnt)
#define USE_ASYNC 1
#else
#define USE_ASYNC 0
#endif

#if USE_ASYNC
typedef int v4i_vs __attribute__((vector_size(16)));
typedef __attribute__((address_space(1))) v4i_vs as1_v4i;
typedef __attribute__((address_space(3))) v4i_vs as3_v4i;
__device__ __forceinline__ void async_g2l_b128(const void* g, void* l) {
  // AS(3) pointers are the low-32-bit LDS offset of the generic address
  // (ISA 10.2).
  __builtin_amdgcn_global_load_async_to_lds_b128(
      (as1_v4i*)(uintptr_t)g, (as3_v4i*)(unsigned)(uintptr_t)l, 0, 0);
}
__device__ __forceinline__ void async_wait0() {
  __builtin_amdgcn_s_wait_asynccnt(0);
}
#else
__device__ __forceinline__ void async_wait0() {}
#endif

// ======================================================================
// Embedding gather: x = tok_emb[ids] + pos_emb
// ======================================================================
__global__ __launch_bounds__(256) void embed_kernel(const int* __restrict__ ids,
    const float* __restrict__ tok, const float* __restrict__ pos,
    float* __restrict__ x)
{
  int i = blockIdx.x * 256 + threadIdx.x;
  int l = i >> 9;
  int d = i & 511;
  x[i] = tok[(size_t)ids[l] * DMODEL + d] + pos[i];
}

// ======================================================================
// LayerNorm over D=512, one row per 256-thread block (8 waves, wave32)
// ======================================================================
__global__ __launch_bounds__(256) void ln_kernel(const float* __restrict__ x,
    const float* __restrict__ g, const float* __restrict__ b,
    float* __restrict__ y)
{
  __shared__ float red[2][8];
  int row = blockIdx.x, tid = threadIdx.x;
  const float* xr = x + (size_t)row * DMODEL;
  float a0 = xr[tid], a1 = xr[tid + 256];
  float s = a0 + a1, sq = a0 * a0 + a1 * a1;
  for (int off = 16; off; off >>= 1) {
    s  += __shfl_xor(s,  off, 32);
    sq += __shfl_xor(sq, off, 32);
  }
  int lane = tid & 31, wv = tid >> 5;
  if (!lane) { red[0][wv] = s; red[1][wv] = sq; }
  __syncthreads();
  float ts = 0.f, tsq = 0.f;
  #pragma unroll
  for (int i = 0; i < 8; ++i) { ts += red[0][i]; tsq += red[1][i]; }
  float mean = ts * (1.f / DMODEL);
  float var  = tsq * (1.f / DMODEL) - mean * mean;
  float rstd = rsqrtf(var + 1e-5f);
  float* yr = y + (size_t)row * DMODEL;
  yr[tid]       = (a0 - mean) * rstd * g[tid]       + b[tid];
  yr[tid + 256] = (a1 - mean) * rstd * g[tid + 256] + b[tid + 256];
}

// ======================================================================
// WMMA fp32 GEMM:  C[M,N] = epilogue( A[M,K] @ B + bias ) (+ residual)
//   BL==0 : B is [K,N] row-major (x @ W)
//   BL==1 : B is [N,K] row-major (x @ W^T), N-bounds guarded (loads clamp,
//           stores predicate), used for logits (N = 50277)
//   ACT==1: exact GELU
// Tiling: 256 thr = 8 waves as 2(M)x4(N); block tile 64x128, K-chunk 64.
// Each wave owns a 32x32 C tile = 2x2 fragments of v_wmma_f32_16x16x4_f32
// (4 WMMA per 2 A-frag + 2 B-frag LDS loads).
// A/B tiles staged memory->LDS with GLOBAL_LOAD_ASYNC_TO_LDS_B128 when the
// toolchain exposes it (s_wait_asynccnt 0 before the barrier).
// ======================================================================
#define BM 64
#define BN 128
#define BK 64

template<int BL, int ACT, bool BIAS, bool RES>
__global__ __launch_bounds__(256) void gemm_kernel(
    const float* __restrict__ A, const float* __restrict__ B,
    const float* __restrict__ bias, const float* __restrict__ res,
    float* __restrict__ C, int M, int N, int K)
{
  __shared__ __align__(16) float As[BM][BK + 4];   // pitch 68 (16B-multiple)
  __shared__ __align__(16) float Bs[BK][BN + 4];   // pitch 132 (16B-multiple)

  const int tid  = threadIdx.x;
  const int lane = tid & 31;
  const int wave = tid >> 5;
  const int wm   = wave >> 2;   // 0..1  (M direction, 32 rows each)
  const int wn   = wave & 3;    // 0..3  (N direction, 32 cols each)
  const int m0   = blockIdx.y * BM;
  const int n0   = blockIdx.x * BN;
  const int lhi  = lane >> 4;   // K-half select (A/B) / M-half select (C)
  const int llo  = lane & 15;

  v8f acc00 = {}, acc01 = {}, acc10 = {}, acc11 = {};

  for (int k0 = 0; k0 < K; k0 += BK) {
    // ---- stage A tile (64x64 f32): 1024 float4, 4 per thread ----
#if USE_ASYNC
    #pragma unroll
    for (int i = 0; i < (BM * BK) / (4 * 256); ++i) {
      int j = tid + i * 256;
      int m  = j >> 4;
      int kk = (j & 15) << 2;
      async_g2l_b128(A + (size_t)(m0 + m) * K + k0 + kk, &As[m][kk]);
    }
#else
    for (int j = tid; j < (BM * BK) / 4; j += 256) {
      int m  = j >> 4;
      int kk = (j & 15) << 2;
      float4 a = *(const float4*)(A + (size_t)(m0 + m) * K + k0 + kk);
      As[m][kk] = a.x; As[m][kk + 1] = a.y; As[m][kk + 2] = a.z; As[m][kk + 3] = a.w;
    }
#endif
    // ---- stage B tile (64x128 f32) ----
    if (BL == 0) {
#if USE_ASYNC
      #pragma unroll
      for (int i = 0; i < (BK * BN) / (4 * 256); ++i) {
        int j = tid + i * 256;
        int k  = j >> 5;
        int nn = (j & 31) << 2;
        async_g2l_b128(B + (size_t)(k0 + k) * N + n0 + nn, &Bs[k][nn]);
      }
#else
      for (int j = tid; j < (BK * BN) / 4; j += 256) {
        int k  = j >> 5;
        int nn = (j & 31) << 2;
        float4 b = *(const float4*)(B + (size_t)(k0 + k) * N + n0 + nn);
        Bs[k][nn] = b.x; Bs[k][nn + 1] = b.y; Bs[k][nn + 2] = b.z; Bs[k][nn + 3] = b.w;
      }
#endif
    } else {
      // B^T staging with transpose; clamp OOB rows (stores are guarded).
      for (int j = tid; j < (BN * BK) / 4; j += 256) {
        int n  = j >> 4;
        int kk = (j & 15) << 2;
        int nc = n0 + n; if (nc > N - 1) nc = N - 1;
        float4 b = *(const float4*)(B + (size_t)nc * K + k0 + kk);
        Bs[kk][n] = b.x; Bs[kk + 1][n] = b.y; Bs[kk + 2][n] = b.z; Bs[kk + 3][n] = b.w;
      }
    }
    async_wait0();
    __syncthreads();

    const int arow0 = wm * 32 + llo;
    const int bcol0 = wn * 32 + llo;
    #pragma unroll
    for (int k4 = 0; k4 < BK; k4 += 4) {
      int ka = k4 + (lhi << 1);
      v2f a0; a0.x = As[arow0][ka];        a0.y = As[arow0][ka + 1];
      v2f a1; a1.x = As[arow0 + 16][ka];   a1.y = As[arow0 + 16][ka + 1];
      v2f b0; b0.x = Bs[ka][bcol0];        b0.y = Bs[ka + 1][bcol0];
      v2f b1; b1.x = Bs[ka][bcol0 + 16];   b1.y = Bs[ka + 1][bcol0 + 16];
      acc00 = __builtin_amdgcn_wmma_f32_16x16x4_f32(false, a0, false, b0,
                                                    (short)0, acc00, false, false);
      acc01 = __builtin_amdgcn_wmma_f32_16x16x4_f32(false, a0, false, b1,
                                                    (short)0, acc01, false, false);
      acc10 = __builtin_amdgcn_wmma_f32_16x16x4_f32(false, a1, false, b0,
                                                    (short)0, acc10, false, false);
      acc11 = __builtin_amdgcn_wmma_f32_16x16x4_f32(false, a1, false, b1,
                                                    (short)0, acc11, false, false);
    }
    __syncthreads();
  }

  // ---- epilogue ----
  const int c0 = n0 + wn * 32 + llo;
  const int c1 = c0 + 16;
  #pragma unroll
  for (int r = 0; r < 8; ++r) {
    int rowA = m0 + wm * 32 + r + (lhi << 3);
    int rowB = rowA + 16;
    size_t baseA = (size_t)rowA * N;
    size_t baseB = (size_t)rowB * N;
    float v00 = acc00[r], v01 = acc01[r], v10 = acc10[r], v11 = acc11[r];
    if (BIAS) {
      float bb0 = (BL == 0 || c0 < N) ? bias[c0] : 0.f;
      float bb1 = (BL == 0 || c1 < N) ? bias[c1] : 0.f;
      v00 += bb0; v10 += bb0; v01 += bb1; v11 += bb1;
    }
    if (ACT == 1) {
      v00 = 0.5f * v00 * (1.0f + erff(v00 * 0.70710678118654752f));
      v01 = 0.5f * v01 * (1.0f + erff(v01 * 0.70710678118654752f));
      v10 = 0.5f * v10 * (1.0f + erff(v10 * 0.70710678118654752f));
      v11 = 0.5f * v11 * (1.0f + erff(v11 * 0.70710678118654752f));
    }
    if (BL == 0 || c0 < N) {
      if (RES) { v00 += res[baseA + c0]; v10 += res[baseB + c0]; }
      C[baseA + c0] = v00; C[baseB + c0] = v10;
    }
    if (BL == 0 || c1 < N) {
      if (RES) { v01 += res[baseA + c1]; v11 += res[baseB + c1]; }
      C[baseA + c1] = v01; C[baseB + c1] = v11;
    }
  }
}

// ======================================================================
// Strided causal windowed attention (WIN=8). One wave per (l, head):
// lane covers dims {lane, lane+32} of DH=64. stride by head group.
// ======================================================================
__global__ __launch_bounds__(256) void attn_kernel(const float* __restrict__ q,
    const float* __restrict__ k, const float* __restrict__ v,
    float* __restrict__ o)
{
  int lane = threadIdx.x & 31;
  int wv   = threadIdx.x >> 5;
  int task = blockIdx.x * 8 + wv;       // L*NHEAD tasks
  int l = task >> 3;
  int h = task & 7;
  int stride = (h < 3) ? 1 : (h < 6) ? 8 : 64;
  int cA = h * 64 + lane;
  int cB = cA + 32;
  size_t rowl = (size_t)l * DMODEL;
  float q0 = q[rowl + cA], q1 = q[rowl + cB];

  float sc[8];
  int   idxs[8];
  #pragma unroll
  for (int w = 0; w < 8; ++w) {
    int idx = l - stride * w;           // uniform across the wave
    idxs[w] = idx;
    float p = -1e30f;
    if (idx >= 0) {
      size_t rw = (size_t)idx * DMODEL;
      float part = q0 * k[rw + cA] + q1 * k[rw + cB];
      for (int off = 16; off; off >>= 1) part += __shfl_xor(part, off, 32);
      p = part * 0.125f;                // DH^-0.5
    }
    sc[w] = p;
  }
  float m = sc[0];
  #pragma unroll
  for (int w = 1; w < 8; ++w) m = fmaxf(m, sc[w]);
  float pw[8], s = 0.f;
  #pragma unroll
  for (int w = 0; w < 8; ++w) {
    pw[w] = (idxs[w] >= 0) ? __expf(sc[w] - m) : 0.f;
    s += pw[w];
  }
  float inv = 1.f / s;
  float o0 = 0.f, o1 = 0.f;
  #pragma unroll
  for (int w = 0; w < 8; ++w) {
    if (idxs[w] >= 0) {
      size_t rw = (size_t)idxs[w] * DMODEL;
      o0 += pw[w] * v[rw + cA];
      o1 += pw[w] * v[rw + cB];
    }
  }
  o[rowl + cA] = o0 * inv;
  o[rowl + cB] = o1 * inv;
}

// ======================================================================
// s4 helpers
// ======================================================================
__global__ __launch_bounds__(256) void gemv_regs_q(const float* __restrict__ regs,
    const float* __restrict__ qw, float* __restrict__ qv)
{
  int j = blockIdx.x * 256 + threadIdx.x;          // < 512
  float acc = 0.f;
  for (int i = 0; i < 768; ++i) acc += regs[i] * qw[(size_t)i * DMODEL + j];
  qv[j] = acc;
}

__global__ __launch_bounds__(256) void s4_score(const float* __restrict__ qv,
    const float* __restrict__ k, float* __restrict__ attn)
{
  int lane = threadIdx.x & 31, wv = threadIdx.x >> 5;
  int l = blockIdx.x * 8 + wv;
  float p = 0.f;
  const float* kr = k + (size_t)l * DMODEL;
  for (int i = lane; i < DMODEL; i += 32) p += qv[i] * kr[i];
  for (int off = 16; off; off >>= 1) p += __shfl_xor(p, off, 32);
  if (!lane) attn[l] = p * 0.044194173824159216f;  // 512^-0.5
}

__global__ __launch_bounds__(1024) void softmax4096_kernel(float* __restrict__ a)
{
  __shared__ float sh[32];
  int tid = threadIdx.x, lane = tid & 31, wv = tid >> 5;
  float v0 = a[tid], v1 = a[tid + 1024], v2 = a[tid + 2048], v3 = a[tid + 3072];
  float m = fmaxf(fmaxf(v0, v1), fmaxf(v2, v3));
  for (int off = 16; off; off >>= 1) m = fmaxf(m, __shfl_xor(m, off, 32));
  if (!lane) sh[wv] = m;
  __syncthreads();
  float gm = -1e30f;
  #pragma unroll
  for (int i = 0; i < 32; ++i) gm = fmaxf(gm, sh[i]);
  __syncthreads();
  float e0 = __expf(v0 - gm), e1 = __expf(v1 - gm);
  float e2 = __expf(v2 - gm), e3 = __expf(v3 - gm);
  float s = e0 + e1 + e2 + e3;
  for (int off = 16; off; off >>= 1) s += __shfl_xor(s, off, 32);
  if (!lane) sh[wv] = s;
  __syncthreads();
  float gs = 0.f;
  #pragma unroll
  for (int i = 0; i < 32; ++i) gs += sh[i];
  float inv = 1.f / gs;
  a[tid] = e0 * inv; a[tid + 1024] = e1 * inv;
  a[tid + 2048] = e2 * inv; a[tid + 3072] = e3 * inv;
}

__global__ __launch_bounds__(256) void s4_summary(const float* __restrict__ w,
    const float* __restrict__ v, float* __restrict__ summary)
{
  int d = blockIdx.x * 256 + threadIdx.x;          // < 512
  float acc = 0.f;
  for (int l = 0; l < LSEQ; ++l) acc += w[l] * v[(size_t)l * DMODEL + d];
  summary[d] = acc;
}

__global__ __launch_bounds__(256) void s4_regs_upd(const float* __restrict__ summary,
    const float* __restrict__ sw, float* __restrict__ regs)
{
  int j = blockIdx.x * 256 + threadIdx.x;          // < 768
  float acc = 0.f;
  for (int d = 0; d < DMODEL; ++d) acc += summary[d] * sw[(size_t)d * 768 + j];
  regs[j] += acc;
}

// ======================================================================
// s3 helpers
// ======================================================================
__global__ __launch_bounds__(256) void col_mean_diff(const float* __restrict__ h,
    const float* __restrict__ x, float* __restrict__ s)
{
  int d = blockIdx.x * 256 + threadIdx.x;          // < 512
  float acc = 0.f;
  for (int l = 0; l < LSEQ; ++l) {
    size_t i = (size_t)l * DMODEL + d;
    acc += h[i] - x[i];
  }
  s[d] = acc * (1.f / LSEQ);
}

__global__ __launch_bounds__(256) void s3_gate(const float* __restrict__ regs,
    const float* __restrict__ s, const float* __restrict__ gw,
    const float* __restrict__ gb, float* __restrict__ gate)
{
  int j = blockIdx.x * 256 + threadIdx.x;          // < 512
  float acc = gb[j];
  for (int i = 0; i < 768; ++i)  acc += regs[i] * gw[(size_t)i * DMODEL + j];
  for (int i = 0; i < 512; ++i)  acc += s[i] * gw[(size_t)(768 + i) * DMODEL + j];
  gate[j] = 1.f / (1.f + __expf(-acc));
}

__global__ __launch_bounds__(256) void s3_apply(float* __restrict__ x,
    const float* __restrict__ h, const float* __restrict__ gate)
{
  size_t i = (size_t)blockIdx.x * 256 + threadIdx.x;
  int d = (int)(i & 511);
  float xv = x[i];
  x[i] = xv + gate[d] * (h[i] - xv);
}

__global__ __launch_bounds__(256) void s3_regs_upd(const float* __restrict__ s,
    const float* __restrict__ wproj, const float* __restrict__ wgw,
    const float* __restrict__ wgb, float* __restrict__ regs, int w0)
{
  __shared__ float red[8];
  int r = blockIdx.x, tid = threadIdx.x;           // r in 0..2, tid < 256
  int g = w0 + r;
  float p = wgw[(size_t)g * DMODEL + tid] * s[tid]
          + wgw[(size_t)g * DMODEL + tid + 256] * s[tid + 256];
  for (int off = 16; off; off >>= 1) p += __shfl_xor(p, off, 32);
  int lane = tid & 31, wv = tid >> 5;
  if (!lane) red[wv] = p;
  __syncthreads();
  float tot = 0.f;
  #pragma unroll
  for (int i = 0; i < 8; ++i) tot += red[i];
  float wg = 1.f / (1.f + __expf(-(tot + wgb[g])));
  float acc = 0.f;
  for (int d = 0; d < DMODEL; ++d)
    acc += s[d] * wproj[((size_t)g * DMODEL + d) * 256 + tid];
  regs[r * 256 + tid] += wg * acc;
}

// ======================================================================
// Host orchestration
// ======================================================================
// Flat input index map (jax pytree sorted-dict order):
//  0 input_ids
//  consolidate[i] base = 1+16*i ; converge[i] base = 49+16*i ; comp layout:
//   +0 ffn.b1 +1 ffn.b2 +2 ffn.ln.b +3 ffn.ln.g +4 ffn.w1 +5 ffn.w2
//   +6 k.b +7 k.w +8 ln.b +9 ln.g +10 o.b +11 o.w +12 q.b +13 q.w +14 v.b +15 v.w
//  81 out_ln.b  82 out_ln.g  83 pos_emb
//  prep: 84 b1, 85 b2, 86 ln.b, 87 ln.g, 88 w1, 89 w2
//  90 reg_init
//  s3: 91 gate_b, 92 gate_w, 93 wgate_b, 94 wgate_w, 95 wproj
//  s4: 96 kw, 97 ln.b, 98 ln.g, 99 qw, 100 sw, 101 vw
//  102 tok_emb

extern "C" void kernel_launch(void* const* d_in, const int* in_sizes, int n_in,
                              void* d_out, int out_size, void* d_ws, size_t ws_size,
                              hipStream_t stream)
{
  if (n_in < 103) return;
  auto F = [&](int i) { return (const float*)d_in[i]; };
  const int*   IDS = (const int*)d_in[0];
  const float* TOK = F(102);
  const float* POS = F(83);
  const int L = LSEQ, Dm = DMODEL, V = 50277;
  const size_t LD = (size_t)L * Dm;

  float* ws = (float*)d_ws;
  float* X   = ws;
  float* CUR = X   + LD;
  float* XN  = CUR + LD;
  float* T0  = XN  + LD;   // q   (aliased as FFN hidden start)
  float* T1  = T0  + LD;   // k
  float* T2  = T1  + LD;   // v
  float* O   = T2  + LD;   // attention out
  float* H1  = T0;         // FFN hidden, up to L*2048 = 4*LD floats
  float* SM  = O + LD;
  float* REGS = SM;          // 768
  float* QV   = REGS + 768;  // 512
  float* ATTW = QV + 512;    // 4096
  float* SVEC = ATTW + 4096; // 512
  float* GATE = SVEC + 512;  // 512
  float* SUMM = GATE + 512;  // 512

  dim3 blk(256);
  auto gemm_grid = [&](int N_) { return dim3((N_ + BN - 1) / BN, L / BM); };

  auto LN = [&](const float* x, const float* g, const float* b, float* y) {
    ln_kernel<<<L, blk, 0, stream>>>(x, g, b, y);
  };
  // C = A@B + bias               (B is [K,N])
  auto GEMM_B = [&](const float* A, const float* B, const float* bias, float* C,
                    int N_, int K_) {
    gemm_kernel<0, 0, true, false><<<gemm_grid(N_), blk, 0, stream>>>(
        A, B, bias, nullptr, C, L, N_, K_);
  };
  // C = gelu(A@B + bias)
  auto GEMM_BG = [&](const float* A, const float* B, const float* bias, float* C,
                     int N_, int K_) {
    gemm_kernel<0, 1, true, false><<<gemm_grid(N_), blk, 0, stream>>>(
        A, B, bias, nullptr, C, L, N_, K_);
  };
  // C = res + A@B + bias
  auto GEMM_BR = [&](const float* A, const float* B, const float* bias,
                     const float* res, float* C, int N_, int K_) {
    gemm_kernel<0, 0, true, true><<<gemm_grid(N_), blk, 0, stream>>>(
        A, B, bias, res, C, L, N_, K_);
  };
  // C = A@B (no bias)
  auto GEMM_P = [&](const float* A, const float* B, float* C, int N_, int K_) {
    gemm_kernel<0, 0, false, false><<<gemm_grid(N_), blk, 0, stream>>>(
        A, B, nullptr, nullptr, C, L, N_, K_);
  };

  // ---- x = tok_emb[ids] + pos_emb ; regs = reg_init ----
  (void)hipMemcpyAsync(REGS, F(90), 768 * sizeof(float),
                       hipMemcpyDeviceToDevice, stream);
  embed_kernel<<<(int)(LD / 256), blk, 0, stream>>>(IDS, TOK, POS, X);

  auto ffn_into = [&](const float* xin, float* xout, int base_b1, int base_b2,
                      int base_lnb, int base_lng, int base_w1, int base_w2, int dff) {
    LN(xin, F(base_lng), F(base_lnb), XN);
    GEMM_BG(XN, F(base_w1), F(base_b1), H1, dff, Dm);
    GEMM_BR(H1, F(base_w2), F(base_b2), xin, xout, Dm, dff);
  };

  auto compressor = [&](float* cur, int base, int dff) {
    // attention sublayer
    LN(cur, F(base + 9), F(base + 8), XN);
    GEMM_B(XN, F(base + 13), F(base + 12), T0, Dm, Dm);   // q
    GEMM_B(XN, F(base + 7),  F(base + 6),  T1, Dm, Dm);   // k
    GEMM_B(XN, F(base + 15), F(base + 14), T2, Dm, Dm);   // v
    attn_kernel<<<L * NHEAD / 8, blk, 0, stream>>>(T0, T1, T2, O);
    GEMM_BR(O, F(base + 11), F(base + 10), cur, cur, Dm, Dm);
    // FFN sublayer (in place on cur)
    ffn_into(cur, cur, base + 0, base + 1, base + 2, base + 3, base + 4, base + 5, dff);
  };

  auto s3_phase = [&](int phase, int it) {
    int hi = it * 3 + phase;
    col_mean_diff<<<2, blk, 0, stream>>>(CUR, X, SVEC);
    s3_gate<<<2, blk, 0, stream>>>(REGS, SVEC,
        F(92) + (size_t)hi * 1280 * Dm, F(91) + (size_t)hi * Dm, GATE);
    s3_regs_upd<<<3, blk, 0, stream>>>(SVEC, F(95), F(94), F(93), REGS, phase * 3);
    s3_apply<<<(int)(LD / 256), blk, 0, stream>>>(X, CUR, GATE);
  };

  for (int it = 0; it < 2; ++it) {
    // ---- s4_intelligence ----
    gemv_regs_q<<<2, blk, 0, stream>>>(REGS, F(99), QV);
    LN(X, F(98), F(97), XN);
    GEMM_P(XN, F(96),  T1, Dm, Dm);                       // k
    GEMM_P(XN, F(101), T2, Dm, Dm);                       // v
    s4_score<<<L / 8, blk, 0, stream>>>(QV, T1, ATTW);
    softmax4096_kernel<<<1, 1024, 0, stream>>>(ATTW);
    s4_summary<<<2, blk, 0, stream>>>(ATTW, T2, SUMM);
    s4_regs_upd<<<3, blk, 0, stream>>>(SUMM, F(100), REGS);

    // ---- prep FFN: CUR = ffn(X) ----
    ffn_into(X, CUR, 84, 85, 86, 87, 88, 89, 1536);
    s3_phase(0, it);

    // ---- converge: 2 compressor layers (DFF=1536) ----
    (void)hipMemcpyAsync(CUR, X, LD * sizeof(float),
                         hipMemcpyDeviceToDevice, stream);
    compressor(CUR, 49, 1536);
    compressor(CUR, 65, 1536);
    s3_phase(1, it);

    // ---- consolidate: 3 compressor layers (DFFC=2048) ----
    (void)hipMemcpyAsync(CUR, X, LD * sizeof(float),
                         hipMemcpyDeviceToDevice, stream);
    compressor(CUR, 1,  2048);
    compressor(CUR, 17, 2048);
    compressor(CUR, 33, 2048);
    s3_phase(2, it);
  }

  // ---- final LN + logits = xn @ tok_emb^T ----
  LN(X, F(82), F(81), XN);
  gemm_kernel<1, 0, false, false><<<gemm_grid(V), blk, 0, stream>>>(
      XN, TOK, nullptr, nullptr, (float*)d_out, L, V, Dm);
}